// IsotonicLayer_13202729468219
// MI455X (gfx1250) — compile-verified
//
#include <hip/hip_runtime.h>
#include <hip/hip_bf16.h>

// IsotonicLayer on MI455X (gfx1250).
//
// Reference: out[b,u] = sigmoid( BW*csum[u,idx] + delta*relu_w[u,idx] + RESIDUE + bias[u] )
// where idx/delta come from bucketizing x[b,u].
//
// Memory-bound: 64 MB in (x) + 64 MB out at 23.3 TB/s ~= 5.5 us floor; params ~32 KB.
// Strategy:
//   K1: build fused table tab[u][b] = { BW*excl_cumsum(relu(w))[u,b] + RESIDUE + bias[u],
//                                       relu(w)[u,b] }  (float2, stride padded to 512)
//       via per-wave (wave32) shuffle scan. Table lives in d_ws (64 KB).
//   K2: stage table into LDS with gfx1250 async global->LDS B128 copies (ASYNCcnt),
//       then grid-stride float4 pass: clip -> idx -> LDS gather (ds_load_b64) -> sigmoid.

#define ISO_UNITS   16
#define ISO_NB      501
#define ISO_NBP     512                  // padded row stride (64 KB total table)
#define ISO_LOWER   (-17.0f)
#define ISO_UPPER   (8.0f)
#define ISO_BW      (0.05f)
#define ISO_RESIDUE (ISO_LOWER - ISO_BW) // -17.05

// ---------------------------------------------------------------------------
// Kernel 1: one block, 512 threads = 16 wave32s; wave w owns unit u = w.
// Exclusive cumsum over 501 buckets via 16 chunks of 32 + carry.
// ---------------------------------------------------------------------------
__global__ void __launch_bounds__(512)
iso_prep_kernel(const float* __restrict__ weights,   // [16, 501]
                const float* __restrict__ bias,      // [16]
                float2* __restrict__ tab)            // [16, 512] in d_ws
{
    const int lane = threadIdx.x & 31;
    const int u    = threadIdx.x >> 5;               // 0..15
    const float cb = ISO_RESIDUE + bias[u];          // fold residue+bias into base

    float carry = 0.0f;
    #pragma unroll 1
    for (int chunk = 0; chunk < 16; ++chunk) {
        const int b = chunk * 32 + lane;
        float wv = 0.0f;
        if (b < ISO_NB) wv = fmaxf(weights[u * ISO_NB + b], 0.0f);   // ReLU

        // Inclusive wave32 scan (Hillis-Steele via shuffles).
        float incl = wv;
        #pragma unroll
        for (int d = 1; d < 32; d <<= 1) {
            float t = __shfl_up(incl, d, 32);
            if (lane >= d) incl += t;
        }
        const float excl = incl - wv;                // exclusive within chunk

        if (b < ISO_NB) {
            float2 e;
            e.x = ISO_BW * (carry + excl) + cb;      // BW*csum + RESIDUE + bias
            e.y = wv;                                // relu(w)
            tab[u * ISO_NBP + b] = e;
        }
        carry += __shfl(incl, 31, 32);               // wave total
    }
}

// ---------------------------------------------------------------------------
// Kernel 2: main elementwise pass. 256 threads/block (8 wave32s), 64 KB LDS.
// ---------------------------------------------------------------------------
__global__ void __launch_bounds__(256)
iso_main_kernel(const float4* __restrict__ x4,       // [B*16/4]
                const float2* __restrict__ tab,      // [16, 512] in d_ws
                float4* __restrict__ out4,           // [B*16/4]
                int n4)
{
    __shared__ float2 stab[ISO_UNITS * ISO_NBP];     // 65536 bytes

    // ---- Stage the 64 KB table with gfx1250 async global->LDS copies. ----
    // GVS addressing: global = SGPR64 base + per-lane VGPR32 offset.
    // 256 threads x 16 iters x 16 B = 65536 B, exact.
    {
        const unsigned lds_base = (unsigned)(size_t)(&stab[0]);
        const void* gbase = (const void*)tab;
        #pragma unroll
        for (int i = 0; i < 16; ++i) {
            const unsigned off      = (unsigned)(i * 256 + threadIdx.x) * 16u;
            const unsigned lds_addr = lds_base + off;
            asm volatile("global_load_async_to_lds_b128 %0, %1, %2"
                         :
                         : "v"(lds_addr), "v"(off), "s"(gbase)
                         : "memory");
        }
        asm volatile("s_wait_asynccnt 0" ::: "memory");
    }
    __syncthreads();

    // ---- Grid-stride over float4 (4 consecutive units of one sample). ----
    const int stride = (int)(gridDim.x * blockDim.x);
    for (int t = (int)(blockIdx.x * blockDim.x + threadIdx.x); t < n4; t += stride) {
        const float4 xv = x4[t];
        const int u0 = (t & 3) * 4;                  // (4t) mod 16

        const float xs[4] = {xv.x, xv.y, xv.z, xv.w};
        float r[4];
        #pragma unroll
        for (int j = 0; j < 4; ++j) {
            // clip (the reference's +-1e-9 epsilons vanish in fp32)
            float xc = fminf(fmaxf(xs[j], ISO_LOWER), ISO_UPPER);
            // bucket index, matching reference op order
            float q  = (xc - ISO_LOWER + ISO_BW) / ISO_BW;
            int idx  = (int)q;                       // truncation, q >= 0
            idx = idx < 0 ? 0 : (idx > ISO_NB - 1 ? ISO_NB - 1 : idx);
            float delta = xc - ISO_LOWER + ISO_BW - (float)idx * ISO_BW;

            // divergent gather from LDS (ds_load_b64)
            const float2 e = stab[(u0 + j) * ISO_NBP + idx];
            const float z = e.x + delta * e.y;       // base + frac + residue + bias

            r[j] = 1.0f / (1.0f + __expf(-z));       // sigmoid
        }
        float4 ov;
        ov.x = r[0]; ov.y = r[1]; ov.z = r[2]; ov.w = r[3];
        out4[t] = ov;
    }
}

// ---------------------------------------------------------------------------
// Host entry
// ---------------------------------------------------------------------------
extern "C" void kernel_launch(void* const* d_in, const int* in_sizes, int n_in,
                              void* d_out, int out_size, void* d_ws, size_t ws_size,
                              hipStream_t stream) {
    const float* x       = (const float*)d_in[0];    // [1e6, 16]
    const float* weights = (const float*)d_in[1];    // [16, 501]
    const float* bias    = (const float*)d_in[2];    // [16]
    float*       out     = (float*)d_out;            // [1e6, 16]
    float2*      tab     = (float2*)d_ws;            // 16*512 float2 = 64 KB

    (void)in_sizes; (void)n_in; (void)ws_size;

    // K1: build fused lookup table (trivial cost).
    iso_prep_kernel<<<1, 512, 0, stream>>>(weights, bias, tab);

    // K2: main pass. 2048 blocks x 256 threads; staging = 2048*64KB = 128 MB
    // of L2-resident reads (table fits in 192 MB L2 trivially).
    const int n4 = out_size / 4;                     // 4,000,000 float4s
    iso_main_kernel<<<2048, 256, 0, stream>>>((const float4*)x, tab,
                                              (float4*)out, n4);
}